// Circuit_16862041604146
// MI455X (gfx1250) — compile-verified
//
#include <hip/hip_runtime.h>

typedef float v2f __attribute__((ext_vector_type(2)));
typedef float v8f __attribute__((ext_vector_type(8)));

#if defined(__HIP_DEVICE_COMPILE__)
#if !__has_builtin(__builtin_amdgcn_wmma_f32_16x16x4_f32)
#error "missing __builtin_amdgcn_wmma_f32_16x16x4_f32 for gfx1250"
#endif
#if __has_builtin(__builtin_amdgcn_global_load_async_to_lds_b128) && \
    __has_builtin(__builtin_amdgcn_global_store_async_from_lds_b128)
#define QS_ASYNC_LDS 1
#endif
#endif
#ifndef QS_ASYNC_LDS
#define QS_ASYNC_LDS 0
#endif

#define QDIM   (1 << 20)   // 2^20 amplitudes
#define QBATCH 8

#if QS_ASYNC_LDS
typedef int v4i __attribute__((vector_size(16)));
typedef v4i __attribute__((address_space(1))) gv4i;   // global int4
typedef v4i __attribute__((address_space(3))) lv4i;   // LDS int4
#endif

// 16-byte global -> LDS copy (async data-mover path when available).
__device__ __forceinline__ void copy16_g2l(float* l, const float* g) {
#if QS_ASYNC_LDS
  __builtin_amdgcn_global_load_async_to_lds_b128(
      (gv4i*)(uintptr_t)g, (lv4i*)(uint32_t)(uintptr_t)l, 0, 0);
#else
  *(float4*)l = *(const float4*)g;
#endif
}
// 16-byte LDS -> global copy (async data-mover path when available).
__device__ __forceinline__ void copy16_l2g(float* g, const float* l) {
#if QS_ASYNC_LDS
  __builtin_amdgcn_global_store_async_from_lds_b128(
      (gv4i*)(uintptr_t)g, (lv4i*)(uint32_t)(uintptr_t)l, 0, 0);
#else
  *(float4*)g = *(const float4*)l;
#endif
}
__device__ __forceinline__ void wait_async0() {
#if QS_ASYNC_LDS
#if __has_builtin(__builtin_amdgcn_s_wait_asynccnt)
  __builtin_amdgcn_s_wait_asynccnt(0);
#else
  asm volatile("s_wait_asynccnt 0" ::: "memory");
#endif
#endif
}

// Wrapper so the HOST compilation pass never sees the amdgcn builtin.
__device__ __forceinline__ v8f wmma_f32_16x16x4(v2f a, v2f b, v8f c) {
#if defined(__HIP_DEVICE_COMPILE__)
  // (neg_a, A, neg_b, B, c_mod, C, reuse_a, reuse_b); F32 WMMA: A/B neg must be 0
  return __builtin_amdgcn_wmma_f32_16x16x4_f32(false, a, false, b, (short)0, c, false, false);
#else
  (void)a; (void)b;
  return c;
#endif
}

struct Cx { float re, im; };
__device__ __forceinline__ Cx cmul(Cx a, Cx b) {
  return { a.re*b.re - a.im*b.im, a.re*b.im + a.im*b.re };
}
__device__ __forceinline__ Cx cadd(Cx a, Cx b) { return { a.re + b.re, a.im + b.im }; }

// Build M = W (x) W (16x16 complex) into LDS, where W = (U1(x)U1) * U2 * (U1(x)U1).
// One thread per element of M (256 threads).
__device__ __forceinline__ void build_M(const float* u1re, const float* u1im,
                                        const float* u2re, const float* u2im,
                                        float* Mre, float* Mim, float* Mneg, int tid) {
  Cx u1[2][2], u2[4][4];
  for (int i = 0; i < 2; ++i)
    for (int j = 0; j < 2; ++j) u1[i][j] = { u1re[i*2+j], u1im[i*2+j] };
  for (int i = 0; i < 4; ++i)
    for (int j = 0; j < 4; ++j) u2[i][j] = { u2re[i*4+j], u2im[i*4+j] };
  Cx A[4][4];   // A = U1 (x) U1 (high bit = first factor)
  for (int a = 0; a < 4; ++a)
    for (int c = 0; c < 4; ++c)
      A[a][c] = cmul(u1[a>>1][c>>1], u1[a&1][c&1]);
  const int g = tid >> 4, h = tid & 15;
  // W[i][j] = sum_{k,l} A[i][k] * U2[k][l] * A[l][j]   (layer3 * layer2 * layer1)
  auto welem = [&](int i, int j) {
    Cx acc = { 0.f, 0.f };
    for (int k = 0; k < 4; ++k)
      for (int l = 0; l < 4; ++l)
        acc = cadd(acc, cmul(cmul(A[i][k], u2[k][l]), A[l][j]));
    return acc;
  };
  const Cx whi = welem(g >> 2, h >> 2);
  const Cx wlo = welem(g & 3, h & 3);
  const Cx m = cmul(whi, wlo);            // M[g][h] = W[g>>2][h>>2] * W[g&3][h&3]
  Mre[tid] = m.re; Mim[tid] = m.im; Mneg[tid] = -m.im;
}

// Per-lane A-matrix fragments (16x4 f32 WMMA layout):
// lanes 0-15: rows 0-15, vgpr0=K0,vgpr1=K1 ; lanes 16-31: rows 0-15, K2/K3.
__device__ __forceinline__ void load_frags(const float* Mre, const float* Mim, const float* Mneg,
                                           v2f aRe[4], v2f aIm[4], v2f aNeg[4], int tid) {
  const int lane = tid & 31, half = lane >> 4, r = lane & 15;
  for (int k = 0; k < 4; ++k) {
    const int c0 = 4*k + 2*half;
    aRe[k]  = v2f{ Mre[r*16 + c0],  Mre[r*16 + c0 + 1]  };
    aIm[k]  = v2f{ Mim[r*16 + c0],  Mim[r*16 + c0 + 1]  };
    aNeg[k] = v2f{ Mneg[r*16 + c0], Mneg[r*16 + c0 + 1] };
  }
}

// Apply M (16x16 complex) along the base-16 digit with local stride (1<<tb)
// over the 4096-element LDS tile. out = M * x   (columns = 256 (hi,lo) combos).
__device__ __forceinline__ void digit_step(float* lre, float* lim,
                                           const v2f aRe[4], const v2f aIm[4], const v2f aNeg[4],
                                           int tb, int tid) {
  const int wave = tid >> 5, lane = tid & 31;
  const int half = lane >> 4, cl = lane & 15;
  const int tmask = (1 << tb) - 1;
  for (int tt = wave * 2; tt < wave * 2 + 2; ++tt) {   // 8 waves x 2 tiles = 256 cols
    const int c    = tt * 16 + cl;                     // this lane's column
    const int base = ((c >> tb) << (tb + 4)) + (c & tmask);
    v8f accRe = {0,0,0,0,0,0,0,0};
    v8f accIm = {0,0,0,0,0,0,0,0};
    for (int k = 0; k < 4; ++k) {
      // B (4x16) fragment: vgpr0 = row 4k+2*half, vgpr1 = next row, col = cl
      const int r0 = 4*k + 2*half;
      const int i0 = base + (r0 << tb), i1 = i0 + (1 << tb);
      const v2f bRe = v2f{ lre[i0], lre[i1] };
      const v2f bIm = v2f{ lim[i0], lim[i1] };
      // complex matmul: re += Mre*Xre + (-Mim)*Xim ; im += Mim*Xre + Mre*Xim
      accRe = wmma_f32_16x16x4(aRe[k],  bRe, accRe);
      accRe = wmma_f32_16x16x4(aNeg[k], bIm, accRe);
      accIm = wmma_f32_16x16x4(aIm[k],  bRe, accIm);
      accIm = wmma_f32_16x16x4(aRe[k],  bIm, accIm);
    }
    // D (16x16) layout: vgpr v, lane L -> row v+8*(L>>4), col L&15
    for (int v = 0; v < 8; ++v) {
      const int row = v + 8*half;
      const int idx = base + (row << tb);
      lre[idx] = accRe[v];
      lim[idx] = accIm[v];
    }
  }
}

// Pass 1: contiguous 4096-amplitude blocks; apply M on base-16 digits 0,1,2.
__global__ __launch_bounds__(256) void qsim_pass1(
    const float* __restrict__ xre, const float* __restrict__ xim,
    const float* __restrict__ u1re, const float* __restrict__ u1im,
    const float* __restrict__ u2re, const float* __restrict__ u2im,
    float* __restrict__ out) {
  __shared__ __align__(16) float lre[4096];
  __shared__ __align__(16) float lim[4096];
  __shared__ float Mre[256], Mim[256], Mneg[256];
  const int tid = threadIdx.x;
  const int b = blockIdx.x >> 8, blk = blockIdx.x & 255;
  const float* sre = xre + (size_t)b * QDIM + (size_t)blk * 4096;
  const float* sim = xim + (size_t)b * QDIM + (size_t)blk * 4096;
  __builtin_prefetch(sre + tid * 16, 0, 0);   // gfx1250 global_prefetch_b8
  __builtin_prefetch(sim + tid * 16, 0, 0);
  // Issue tile staging (async data movers when available), overlap with build_M.
  for (int it = 0; it < 4; ++it) {
    const int f = (tid + it * 256) * 4;
    copy16_g2l(&lre[f], sre + f);
    copy16_g2l(&lim[f], sim + f);
  }
  build_M(u1re, u1im, u2re, u2im, Mre, Mim, Mneg, tid);
  wait_async0();
  __syncthreads();
  v2f aRe[4], aIm[4], aNeg[4];
  load_frags(Mre, Mim, Mneg, aRe, aIm, aNeg, tid);
  digit_step(lre, lim, aRe, aIm, aNeg, 0, tid); __syncthreads();
  digit_step(lre, lim, aRe, aIm, aNeg, 4, tid); __syncthreads();
  digit_step(lre, lim, aRe, aIm, aNeg, 8, tid); __syncthreads();
  float* ore = out + (size_t)b * QDIM + (size_t)blk * 4096;
  float* oim = out + (size_t)(QBATCH + b) * QDIM + (size_t)blk * 4096;
  for (int it = 0; it < 4; ++it) {
    const int f = (tid + it * 256) * 4;
    copy16_l2g(ore + f, &lre[f]);
    copy16_l2g(oim + f, &lim[f]);
  }
  wait_async0();
}

// Pass 2: tiles = (digit4 x digit3 x 16 contiguous); apply M on digits 3,4 in place.
__global__ __launch_bounds__(256) void qsim_pass2(
    const float* __restrict__ u1re, const float* __restrict__ u1im,
    const float* __restrict__ u2re, const float* __restrict__ u2im,
    float* __restrict__ out) {
  __shared__ __align__(16) float lre[4096];
  __shared__ __align__(16) float lim[4096];
  __shared__ float Mre[256], Mim[256], Mneg[256];
  const int tid = threadIdx.x;
  const int b = blockIdx.x >> 8, blk = blockIdx.x & 255;
  float* pre = out + (size_t)b * QDIM;
  float* pim = out + (size_t)(QBATCH + b) * QDIM;
  // local i = e4*256 + e3*16 + j  <->  global n = e4*65536 + e3*4096 + blk*16 + j
  for (int it = 0; it < 4; ++it) {
    const int f  = (tid + it * 256) * 4;
    const int j  = f & 15, e3 = (f >> 4) & 15, e4 = f >> 8;
    const int n  = (e4 << 16) + (e3 << 12) + blk * 16 + j;
    copy16_g2l(&lre[f], pre + n);
    copy16_g2l(&lim[f], pim + n);
  }
  build_M(u1re, u1im, u2re, u2im, Mre, Mim, Mneg, tid);
  wait_async0();
  __syncthreads();
  v2f aRe[4], aIm[4], aNeg[4];
  load_frags(Mre, Mim, Mneg, aRe, aIm, aNeg, tid);
  digit_step(lre, lim, aRe, aIm, aNeg, 4, tid); __syncthreads();   // digit 3
  digit_step(lre, lim, aRe, aIm, aNeg, 8, tid); __syncthreads();   // digit 4
  for (int it = 0; it < 4; ++it) {
    const int f  = (tid + it * 256) * 4;
    const int j  = f & 15, e3 = (f >> 4) & 15, e4 = f >> 8;
    const int n  = (e4 << 16) + (e3 << 12) + blk * 16 + j;
    copy16_l2g(pre + n, &lre[f]);
    copy16_l2g(pim + n, &lim[f]);
  }
  wait_async0();
}

extern "C" void kernel_launch(void* const* d_in, const int* in_sizes, int n_in,
                              void* d_out, int out_size, void* d_ws, size_t ws_size,
                              hipStream_t stream) {
  (void)in_sizes; (void)n_in; (void)d_ws; (void)ws_size; (void)out_size;
  const float* xre  = (const float*)d_in[0];
  const float* xim  = (const float*)d_in[1];
  const float* u1re = (const float*)d_in[2];
  const float* u1im = (const float*)d_in[3];
  const float* u2re = (const float*)d_in[4];
  const float* u2im = (const float*)d_in[5];
  float* out = (float*)d_out;
  // 8 batches x 256 blocks of 4096 amplitudes each
  qsim_pass1<<<dim3(QBATCH * 256), dim3(256), 0, stream>>>(xre, xim, u1re, u1im, u2re, u2im, out);
  qsim_pass2<<<dim3(QBATCH * 256), dim3(256), 0, stream>>>(u1re, u1im, u2re, u2im, out);
}